// GCNGraphConvLayer_12240656794081
// MI455X (gfx1250) — compile-verified
//
#include <hip/hip_runtime.h>
#include <math.h>

// ---------------------------------------------------------------------------
// R-GCN layer for MI455X (gfx1250, wave32).
// - Scatter stage is L2-resident (192MB L2 >> ~103MB working set).
// - GEMMs use V_WMMA_F32_16X16X4_F32 (exact fp32, matches reference).
// - Weights pre-transposed so both A and B fragments are single b64 loads.
// - deg -> 1/max(deg,1) precomputed once per relation; GEMM templated on
//   HAS_SCALE so the tile epilogue is branch-free and clause-able.
// ---------------------------------------------------------------------------

typedef float v2f __attribute__((ext_vector_type(2)));
typedef float v8f __attribute__((ext_vector_type(8)));

#define DIM 256

// ----------------------------- zero fill (b128) ----------------------------
__global__ void zero_f32_v4(float4* __restrict__ p, int n4) {
  int i = blockIdx.x * blockDim.x + threadIdx.x;
  int stride = gridDim.x * blockDim.x;
  float4 z = {0.0f, 0.0f, 0.0f, 0.0f};
  for (; i < n4; i += stride) p[i] = z;
}

__global__ void zero_f32(float* __restrict__ p, int n) {
  int i = blockIdx.x * blockDim.x + threadIdx.x;
  int stride = gridDim.x * blockDim.x;
  for (; i < n; i += stride) p[i] = 0.0f;
}

// --------------------- deg -> 1/max(deg,1), in place -----------------------
__global__ void invdeg_f32(float* __restrict__ deg, int n) {
  int i = blockIdx.x * blockDim.x + threadIdx.x;
  int stride = gridDim.x * blockDim.x;
  for (; i < n; i += stride) deg[i] = 1.0f / fmaxf(deg[i], 1.0f);
}

// --------------------- 256x256 transpose (LDS tiled) -----------------------
// out[n*256+k] = in[k*256+n], for gridDim.z matrices.
__global__ void transpose256(const float* __restrict__ in,
                             float* __restrict__ out) {
  __shared__ float tile[16][17];          // +1 pad: no LDS bank conflicts
  const float* I = in  + (size_t)blockIdx.z * DIM * DIM;
  float*       O = out + (size_t)blockIdx.z * DIM * DIM;
  int k0 = blockIdx.x * 16;
  int n0 = blockIdx.y * 16;
  int tx = threadIdx.x, ty = threadIdx.y;
  tile[ty][tx] = I[(size_t)(k0 + ty) * DIM + (n0 + tx)];   // coalesced read
  __syncthreads();
  O[(size_t)(n0 + ty) * DIM + (k0 + tx)] = tile[tx][ty];   // coalesced write
}

// --------------------------- edge scatter ----------------------------------
// One wave per edge iteration: 32 lanes x 8 contiguous floats = 256-dim row.
// msg[dst] += x[src]; deg[dst] += 1. Atomics land in L2 (whole msg fits).
__global__ void rgcn_scatter(const float* __restrict__ x,
                             const long long* __restrict__ src,
                             const long long* __restrict__ dst,
                             float* __restrict__ msg,
                             float* __restrict__ deg,
                             int nEdges) {
  int tid    = blockIdx.x * blockDim.x + threadIdx.x;
  int lane   = threadIdx.x & 31;
  int wid    = tid >> 5;
  int nwaves = (gridDim.x * blockDim.x) >> 5;

  for (int e = wid; e < nEdges; e += nwaves) {
    int s = (int)src[e];
    int d = (int)dst[e];

    // prefetch the next iteration's source row into cache (global_prefetch_b8)
    int e2 = e + nwaves;
    if (e2 < nEdges) {
      int s2 = (int)src[e2];
      __builtin_prefetch(x + (size_t)s2 * DIM + lane * 8, 0, 0);
    }

    const float4* xr = (const float4*)(x + (size_t)s * DIM + lane * 8);
    float4 a0 = xr[0];
    float4 a1 = xr[1];

    float* m = msg + (size_t)d * DIM + lane * 8;
    __hip_atomic_fetch_add(m + 0, a0.x, __ATOMIC_RELAXED, __HIP_MEMORY_SCOPE_AGENT);
    __hip_atomic_fetch_add(m + 1, a0.y, __ATOMIC_RELAXED, __HIP_MEMORY_SCOPE_AGENT);
    __hip_atomic_fetch_add(m + 2, a0.z, __ATOMIC_RELAXED, __HIP_MEMORY_SCOPE_AGENT);
    __hip_atomic_fetch_add(m + 3, a0.w, __ATOMIC_RELAXED, __HIP_MEMORY_SCOPE_AGENT);
    __hip_atomic_fetch_add(m + 4, a1.x, __ATOMIC_RELAXED, __HIP_MEMORY_SCOPE_AGENT);
    __hip_atomic_fetch_add(m + 5, a1.y, __ATOMIC_RELAXED, __HIP_MEMORY_SCOPE_AGENT);
    __hip_atomic_fetch_add(m + 6, a1.z, __ATOMIC_RELAXED, __HIP_MEMORY_SCOPE_AGENT);
    __hip_atomic_fetch_add(m + 7, a1.w, __ATOMIC_RELAXED, __HIP_MEMORY_SCOPE_AGENT);

    if (lane == 0)
      __hip_atomic_fetch_add(deg + d, 1.0f, __ATOMIC_RELAXED, __HIP_MEMORY_SCOPE_AGENT);
  }
}

// --------------------------- WMMA fp32 GEMM --------------------------------
// out[nRows x 256] += scale[row] * (A[nRows x 256] @ Wt^T)
// Wt is the TRANSPOSED weight [N x K]; HAS_SCALE=false -> scale 1 (self-loop).
// One wave computes one 16x16 tile via 64x V_WMMA_F32_16X16X4_F32.
// Block = 256 threads = 8 waves = 16 rows x 128 cols; gridDim.y=2 covers 256.
//
// A (16x4 f32) lane layout (ISA 7.12.2): lanes 0-15 hold row M=lane, K={0,1};
// lanes 16-31 hold row M=lane-16, K={2,3}.  B mirrored with N in place of M.
// C/D: VGPR v holds M=v (lanes 0-15) / M=v+8 (lanes 16-31), N=lane&15.
// Inner loop is pure {b64 A, b64 B, wmma}; epilogue is branch-free.
template <bool HAS_SCALE>
__global__ void rgcn_wmma_gemm_acc(const float* __restrict__ A,
                                   const float* __restrict__ scale,
                                   const float* __restrict__ Wt,
                                   float* __restrict__ out,
                                   int nRows) {
  int lane = threadIdx.x & 31;
  int wave = threadIdx.x >> 5;
  int lidx = lane & 15;
  int half = lane >> 4;   // 0: K={0,1}/M rows 0-7 ; 1: K={2,3}/M rows 8-15

  int row0 = blockIdx.x * 16;
  if (row0 >= nRows) return;
  int col0 = (blockIdx.y * 8 + wave) * 16;

  const float* Arow  = A  + (size_t)(row0 + lidx) * DIM;   // this lane's A row
  const float* WtCol = Wt + (size_t)(col0 + lidx) * DIM;   // this lane's B col

  v8f acc = {};
#pragma unroll 8
  for (int k = 0; k < DIM; k += 4) {
    int ka = k + half * 2;
    v2f a = *(const v2f*)(Arow + ka);    // 8B load: A[row][ka..ka+1]
    v2f b = *(const v2f*)(WtCol + ka);   // 8B load: W[ka..ka+1][col]
    acc = __builtin_amdgcn_wmma_f32_16x16x4_f32(
        /*neg_a=*/false, a, /*neg_b=*/false, b,
        /*c_mod=*/(short)0, acc, /*reuse_a=*/false, /*reuse_b=*/false);
  }

  int col = col0 + lidx;

  // gather the 8 row scales up-front (broadcast loads, clause-able)
  float sc[8];
#pragma unroll
  for (int v = 0; v < 8; ++v)
    sc[v] = HAS_SCALE ? scale[row0 + half * 8 + v] : 1.0f;

#pragma unroll
  for (int v = 0; v < 8; ++v) {
    int row = row0 + half * 8 + v;
    out[(size_t)row * DIM + col] += sc[v] * acc[v];  // exclusive tile -> RMW
  }
}

// ------------------------------ epilogue -----------------------------------
__global__ void rgcn_epilogue(float* __restrict__ out,
                              const float* __restrict__ bias, int n) {
  int i = blockIdx.x * blockDim.x + threadIdx.x;
  int stride = gridDim.x * blockDim.x;
  for (; i < n; i += stride) out[i] = tanhf(out[i] + bias[i & (DIM - 1)]);
}

// ---------------------------------------------------------------------------
extern "C" void kernel_launch(void* const* d_in, const int* in_sizes, int n_in,
                              void* d_out, int out_size, void* d_ws, size_t ws_size,
                              hipStream_t stream) {
  const float*     x      = (const float*)d_in[0];
  const long long* edges  = (const long long*)d_in[1];   // int64 [R,2,E]
  const float*     weight = (const float*)d_in[2];       // [R,256,256]
  const float*     loop_w = (const float*)d_in[3];       // [256,256]
  const float*     bias   = (const float*)d_in[4];       // [256]

  const int N = in_sizes[0] / DIM;                 // 50000 nodes
  const int R = in_sizes[2] / (DIM * DIM);         // 4 relations
  const int E = in_sizes[1] / (R * 2);             // 400000 edges/relation
  const int nOut = N * DIM;

  // workspace layout: msg [N*256] | deg [N] | Wt [(R+1)*256*256]
  float* msg = (float*)d_ws;
  float* deg = msg + (size_t)N * DIM;
  float* Wt  = deg + N;
  float* outf = (float*)d_out;

  // transpose all weight matrices once (tiny, L2-resident afterwards)
  transpose256<<<dim3(16, 16, R), dim3(16, 16), 0, stream>>>(weight, Wt);
  transpose256<<<dim3(16, 16, 1), dim3(16, 16), 0, stream>>>(
      loop_w, Wt + (size_t)R * DIM * DIM);

  zero_f32_v4<<<512, 256, 0, stream>>>((float4*)outf, nOut / 4);

  dim3 gemm_grid((N + 15) / 16, 2);
  for (int r = 0; r < R; ++r) {
    zero_f32_v4<<<512, 256, 0, stream>>>((float4*)msg, nOut / 4);
    zero_f32<<<64, 256, 0, stream>>>(deg, N);

    const long long* src = edges + (size_t)r * 2 * E;
    const long long* dst = src + E;
    rgcn_scatter<<<2048, 256, 0, stream>>>(x, src, dst, msg, deg, E);

    invdeg_f32<<<64, 256, 0, stream>>>(deg, N);    // deg := 1/max(deg,1)

    rgcn_wmma_gemm_acc<true><<<gemm_grid, 256, 0, stream>>>(
        msg, deg, Wt + (size_t)r * DIM * DIM, outf, N);
  }

  // self-loop: out += x @ loop_weight  (scale = 1, branch-free instantiation)
  rgcn_wmma_gemm_acc<false><<<gemm_grid, 256, 0, stream>>>(
      x, nullptr, Wt + (size_t)R * DIM * DIM, outf, N);

  rgcn_epilogue<<<512, 256, 0, stream>>>(outf, bias, nOut);
}